// ModelGraphCoordinationNet_41248865910880
// MI455X (gfx1250) — compile-verified
//
#include <hip/hip_runtime.h>
#include <math.h>

typedef __attribute__((ext_vector_type(2))) float v2f;
typedef __attribute__((ext_vector_type(8))) float v8f;

#define NN 50000      // nodes
#define NE 360000     // edges (without self loops)
#define NG 2000       // graphs
#define FD 222        // feature dim
#define FP 224        // padded feature dim (multiple of 16, of 4)

static inline int cdiv(long a, long b) { return (int)((a + b - 1) / b); }

// ---------------- feature assembly: X[i, 0:224] ----------------
__global__ void k_build_x(const int* __restrict__ el, const int* __restrict__ ox,
                          const int* __restrict__ geo, const float* __restrict__ ang,
                          const float* __restrict__ embE, const float* __restrict__ embO,
                          const float* __restrict__ embG, float* __restrict__ X) {
  long t = (long)blockIdx.x * blockDim.x + threadIdx.x;
  if (t >= (long)NN * FP) return;
  int i = (int)(t / FP), f = (int)(t % FP);
  float v;
  if (f < 200)      v = embE[el[i] * 200 + f];
  else if (f < 210) v = embO[ox[i] * 10 + (f - 200)];
  else if (f < 220) v = embG[geo[i] * 10 + (f - 210)];
  else if (f < 222) v = ang[i * 2 + (f - 220)];
  else              v = 0.f;
  X[t] = v;
}

// ---------------- degree / norm ----------------
__global__ void k_deg_init(float* deg) {
  int i = blockIdx.x * blockDim.x + threadIdx.x;
  if (i < NN) deg[i] = 1.0f;  // self loop
}
__global__ void k_deg_edges(const int* __restrict__ dst, float* deg) {
  int e = blockIdx.x * blockDim.x + threadIdx.x;
  if (e < NE) atomicAdd(&deg[dst[e]], 1.0f);
}
__global__ void k_deg_fin(float* deg) {
  int i = blockIdx.x * blockDim.x + threadIdx.x;
  if (i < NN) deg[i] = rsqrtf(deg[i]);  // deg >= 1 always
}

// ---------------- generic zero-pad copy ----------------
__global__ void k_pad(const float* __restrict__ src, int R, int C,
                      float* __restrict__ dst, int Rp, int Cp) {
  long t = (long)blockIdx.x * blockDim.x + threadIdx.x;
  if (t >= (long)Rp * Cp) return;
  int r = (int)(t / Cp), c = (int)(t % Cp);
  dst[t] = (r < R && c < C) ? src[r * C + c] : 0.f;
}

// ======== WMMA fp32 GEMM, 2 N-tiles per wave: C = act(A@B + bias) ========
// Each wave computes a 16x32 strip: two independent accumulator chains,
// A fragments reused for both tiles. Requires N % 32 == 0.
__global__ void k_gemm_wmma2(const float* __restrict__ A, int lda,
                             const float* __restrict__ B, int ldb,
                             const float* __restrict__ bias,
                             float* __restrict__ C, int ldc,
                             int Mtiles, int Npairs, int K, int act) {
  int wid = threadIdx.x >> 5;
  int strip = blockIdx.x * (blockDim.x >> 5) + wid;
  if (strip >= Mtiles * Npairs) return;   // wave-uniform: EXEC stays all-1s
  int lane = threadIdx.x & 31;
  int rc = lane & 15;      // A row within tile == B/C col within tile
  int half = lane >> 4;    // 0 or 1
  int tm = strip / Npairs, tp = strip % Npairs;

  const float* Ar = A + (long)(tm * 16 + rc) * lda;  // A: lane -> row M
  const float* B0 = B + tp * 32 + rc;                // B: lane -> col N
  const float* B1 = B0 + 16;

  v8f acc0 = {}, acc1 = {};
  for (int k = 0; k < K; k += 4) {
    int ka = k + 2 * half;
    v2f a, b0, b1;
    a.x  = Ar[ka];                     // A VGPR0: K = ka
    a.y  = Ar[ka + 1];                 // A VGPR1: K = ka+1
    b0.x = B0[(long)ka * ldb];
    b0.y = B0[(long)(ka + 1) * ldb];
    b1.x = B1[(long)ka * ldb];
    b1.y = B1[(long)(ka + 1) * ldb];
    acc0 = __builtin_amdgcn_wmma_f32_16x16x4_f32(
        false, a, false, b0, (short)0, acc0, false, false);
    acc1 = __builtin_amdgcn_wmma_f32_16x16x4_f32(
        false, a, false, b1, (short)0, acc1, false, false);
  }

  int col0 = tp * 32 + rc;
  float bv0 = bias ? bias[col0]      : 0.f;
  float bv1 = bias ? bias[col0 + 16] : 0.f;
  #pragma unroll
  for (int j = 0; j < 8; j++) {
    int row = tm * 16 + half * 8 + j;  // D: VGPR j -> M=j (lanes 0-15), M=8+j (16-31)
    float v0 = acc0[j] + bv0;
    float v1 = acc1[j] + bv1;
    if (act) {
      v0 = v0 > 0.f ? v0 : (expf(v0) - 1.f);
      v1 = v1 > 0.f ? v1 : (expf(v1) - 1.f);
    }
    C[(long)row * ldc + col0]      = v0;
    C[(long)row * ldc + col0 + 16] = v1;
  }
}

// ---- 1-tile variant (for N == 16 only: final head layer) ----
__global__ void k_gemm_wmma1(const float* __restrict__ A, int lda,
                             const float* __restrict__ B, int ldb,
                             const float* __restrict__ bias,
                             float* __restrict__ C, int ldc,
                             int Mtiles, int K) {
  int wid = threadIdx.x >> 5;
  int tile = blockIdx.x * (blockDim.x >> 5) + wid;
  if (tile >= Mtiles) return;
  int lane = threadIdx.x & 31;
  int rc = lane & 15, half = lane >> 4;

  const float* Ar = A + (long)(tile * 16 + rc) * lda;
  const float* Bc = B + rc;

  v8f acc = {};
  for (int k = 0; k < K; k += 4) {
    int ka = k + 2 * half;
    v2f a, b;
    a.x = Ar[ka];  a.y = Ar[ka + 1];
    b.x = Bc[(long)ka * ldb];
    b.y = Bc[(long)(ka + 1) * ldb];
    acc = __builtin_amdgcn_wmma_f32_16x16x4_f32(
        false, a, false, b, (short)0, acc, false, false);
  }
  float bv = bias[rc];
  #pragma unroll
  for (int j = 0; j < 8; j++) {
    int row = tile * 16 + half * 8 + j;
    C[(long)row * ldc + rc] = acc[j] + bv;
  }
}

// ---------------- edge scatter: Y[dst] += dinv[s]*dinv[d] * H[src] --------
// One wave per edge; self loops appended as edges [NE, NE+NN).
__global__ void k_scatter(const float* __restrict__ H, float* __restrict__ Y,
                          const int* __restrict__ srcI, const int* __restrict__ dstI,
                          const float* __restrict__ dinv) {
  int wid = threadIdx.x >> 5;
  long e = (long)blockIdx.x * (blockDim.x >> 5) + wid;
  if (e >= (long)NE + NN) return;
  int s, d;
  if (e < NE) { s = srcI[e]; d = dstI[e]; }
  else        { s = d = (int)(e - NE); }
  float w = dinv[s] * dinv[d];
  int lane = threadIdx.x & 31;
  const float* hs = H + (long)s * FP;
  float* yd = Y + (long)d * FP;
  for (int f = lane; f < FD; f += 32)
    atomicAdd(&yd[f], w * hs[f]);
}

// ---------------- bias + ELU over node features ----------------
__global__ void k_bias_elu(float* __restrict__ X, const float* __restrict__ b) {
  long t = (long)blockIdx.x * blockDim.x + threadIdx.x;
  if (t >= (long)NN * FP) return;
  int f = (int)(t % FP);
  float v = X[t] + b[f];          // padded cols: 0 + 0 -> elu(0)=0
  X[t] = v > 0.f ? v : (expf(v) - 1.f);
}

// ---------------- global mean pool ----------------
__global__ void k_pool(const float* __restrict__ X, const int* __restrict__ batch,
                       float* __restrict__ sums, float* __restrict__ cnt) {
  int wid = threadIdx.x >> 5;
  int i = blockIdx.x * (blockDim.x >> 5) + wid;
  if (i >= NN) return;
  int g = batch[i];
  int lane = threadIdx.x & 31;
  if (lane == 0) atomicAdd(&cnt[g], 1.0f);
  const float* xi = X + (long)i * FP;
  float* sg = sums + (long)g * FP;
  for (int f = lane; f < FP; f += 32)
    atomicAdd(&sg[f], xi[f]);
}
__global__ void k_pool_div(float* __restrict__ sums, const float* __restrict__ cnt) {
  long t = (long)blockIdx.x * blockDim.x + threadIdx.x;
  if (t >= (long)NG * FP) return;
  float c = cnt[t / FP];
  sums[t] /= (c > 1.f ? c : 1.f);
}

// ---------------- final column extract ----------------
__global__ void k_col0(const float* __restrict__ o16, float* __restrict__ out) {
  int g = blockIdx.x * blockDim.x + threadIdx.x;
  if (g < NG) out[g] = o16[g * 16];
}

extern "C" void kernel_launch(void* const* d_in, const int* in_sizes, int n_in,
                              void* d_out, int out_size, void* d_ws, size_t ws_size,
                              hipStream_t stream) {
  (void)in_sizes; (void)n_in; (void)out_size; (void)ws_size;
  const int*   elements = (const int*)d_in[0];
  const int*   oxid     = (const int*)d_in[1];
  const int*   geom     = (const int*)d_in[2];
  const float* angles   = (const float*)d_in[3];
  const int*   eidx     = (const int*)d_in[4];
  const int*   batch    = (const int*)d_in[5];
  const float* embE     = (const float*)d_in[6];
  const float* embO     = (const float*)d_in[7];
  const float* embG     = (const float*)d_in[8];
  const float* W[3]  = {(const float*)d_in[9],  (const float*)d_in[11], (const float*)d_in[13]};
  const float* b[3]  = {(const float*)d_in[10], (const float*)d_in[12], (const float*)d_in[14]};
  const float* Wd1 = (const float*)d_in[15]; const float* bd1 = (const float*)d_in[16];
  const float* Wd2 = (const float*)d_in[17]; const float* bd2 = (const float*)d_in[18];
  const float* Wd3 = (const float*)d_in[19]; const float* bd3 = (const float*)d_in[20];
  const int* srcI = eidx;
  const int* dstI = eidx + NE;

  // ---- workspace carve-up (256B aligned) ----
  char* p = (char*)d_ws;
  auto take = [&](long nfloats) {
    float* r = (float*)p;
    p += ((nfloats * 4 + 255) / 256) * 256;
    return r;
  };
  float* X    = take((long)NN * FP);
  float* H    = take((long)NN * FP);
  float* dinv = take(NN);
  float* Wp[3], *bp[3];
  for (int l = 0; l < 3; l++) { Wp[l] = take((long)FP * FP); bp[l] = take(FP); }
  float* Wd1p  = take((long)FP * 512);
  float* Wd3p  = take(128 * 16);
  float* bd3p  = take(16);
  float* sums  = take((long)NG * FP);
  float* cnt   = take(NG);
  float* h1    = take((long)NG * 512);
  float* h2    = take((long)NG * 128);
  float* o16   = take((long)NG * 16);

  const int T = 256;

  // features
  k_build_x<<<cdiv((long)NN * FP, T), T, 0, stream>>>(elements, oxid, geom, angles,
                                                      embE, embO, embG, X);
  // degree norm
  k_deg_init<<<cdiv(NN, T), T, 0, stream>>>(dinv);
  k_deg_edges<<<cdiv(NE, T), T, 0, stream>>>(dstI, dinv);
  k_deg_fin<<<cdiv(NN, T), T, 0, stream>>>(dinv);

  // pad parameters
  for (int l = 0; l < 3; l++) {
    k_pad<<<cdiv((long)FP * FP, T), T, 0, stream>>>(W[l], FD, FD, Wp[l], FP, FP);
    k_pad<<<cdiv(FP, T), T, 0, stream>>>(b[l], 1, FD, bp[l], 1, FP);
  }
  k_pad<<<cdiv((long)FP * 512, T), T, 0, stream>>>(Wd1, FD, 512, Wd1p, FP, 512);
  k_pad<<<cdiv(128 * 16, T), T, 0, stream>>>(Wd3, 128, 1, Wd3p, 128, 16);
  k_pad<<<1, T, 0, stream>>>(bd3, 1, 1, bd3p, 1, 16);

  // ---- 3 GCN layers ----
  const int Mt = NN / 16;        // 3125
  const int Np = FP / 32;        // 7 N-pairs
  for (int l = 0; l < 3; l++) {
    // H = X @ Wp[l]
    k_gemm_wmma2<<<cdiv((long)Mt * Np, 8), T, 0, stream>>>(
        X, FP, Wp[l], FP, nullptr, H, FP, Mt, Np, FP, 0);
    // X = 0 ; X[dst] += norm * H[src]
    hipMemsetAsync(X, 0, (size_t)NN * FP * 4, stream);
    k_scatter<<<cdiv((long)NE + NN, 8), T, 0, stream>>>(H, X, srcI, dstI, dinv);
    // X = elu(X + b)
    k_bias_elu<<<cdiv((long)NN * FP, T), T, 0, stream>>>(X, bp[l]);
  }

  // ---- mean pool ----
  hipMemsetAsync(sums, 0, (size_t)NG * FP * 4, stream);
  hipMemsetAsync(cnt, 0, (size_t)NG * 4, stream);
  k_pool<<<cdiv(NN, 8), T, 0, stream>>>(X, batch, sums, cnt);
  k_pool_div<<<cdiv((long)NG * FP, T), T, 0, stream>>>(sums, cnt);

  // ---- dense head (WMMA, fused bias+ELU) ----
  const int Gt = NG / 16;        // 125
  k_gemm_wmma2<<<cdiv((long)Gt * 16, 8), T, 0, stream>>>(
      sums, FP, Wd1p, 512, bd1, h1, 512, Gt, 16, FP, 1);
  k_gemm_wmma2<<<cdiv((long)Gt * 4, 8), T, 0, stream>>>(
      h1, 512, Wd2, 128, bd2, h2, 128, Gt, 4, 512, 1);
  k_gemm_wmma1<<<cdiv(Gt, 8), T, 0, stream>>>(
      h2, 128, Wd3p, 16, bd3p, o16, 16, Gt, 128);

  k_col0<<<cdiv(NG, T), T, 0, stream>>>(o16, (float*)d_out);
}